// GaborBasis_18562848653620
// MI455X (gfx1250) — compile-verified
//
#include <hip/hip_runtime.h>

// ---------------- problem constants ----------------
#define R_   1024   // anchors
#define D_   256    // feature dim
#define BT_  16384  // B*T rows
#define MT   16     // rows (M) per workgroup
#define NC   64     // anchors (N) per chunk  -> 4 WMMA N-tiles
#define LDP  260    // padded row stride (floats) for z/mu/K LDS tiles (16B-aligned, conflict-free)
#define WCP  17     // padded stride for wc tile (conflict-free scatter + coalesced read)
#define NTHREADS 256

typedef __attribute__((ext_vector_type(2))) float v2f;
typedef __attribute__((ext_vector_type(4))) float v4f;
typedef __attribute__((ext_vector_type(8))) float v8f;

// LDS layout (floats):
//  zt   : MT*LDP        = 4160
//  muc  : NC*LDP        = 16640
//  kc   : NC*LDP        = 16640
//  wc   : R_*WCP        = 17408
//  accb : 8*256         = 2048
//  zsq  : 16, rsum: 16, invs: 16
#define SMEM_FLOATS (MT*LDP + 2*(NC*LDP) + R_*WCP + 8*256 + 48)
#define SMEM_BYTES  (SMEM_FLOATS * 4)

// wait for this wave's outstanding async global<->LDS ops
#define ASYNC_WAIT0() asm volatile("s_wait_asynccnt 0x0" ::: "memory")

// ---------------- per-anchor constants: musq[r] = sum_d mu^2, i2s[r] = 0.5/sigma^2 ----------------
__global__ __launch_bounds__(NTHREADS)
void gabor_precompute(const float* __restrict__ mu, const float* __restrict__ log_sigma,
                      float* __restrict__ musq, float* __restrict__ i2s) {
    const int wid  = (blockIdx.x * NTHREADS + threadIdx.x) >> 5;  // one wave per anchor row
    const int lane = threadIdx.x & 31;
    if (wid >= R_) return;
    const float* row = mu + (size_t)wid * D_;
    float s = 0.0f;
    #pragma unroll
    for (int j = lane; j < D_; j += 32) { float v = row[j]; s += v * v; }
    #pragma unroll
    for (int off = 16; off > 0; off >>= 1) s += __shfl_xor(s, off);
    if (lane == 0) {
        musq[wid] = s;
        float sg = __expf(log_sigma[wid]);
        i2s[wid] = 0.5f / (sg * sg + 1e-8f);
    }
}

// ---------------- async DMA staging of one mu/K chunk (64 rows x 256 f32 each) ----------------
// Per-lane GLOBAL_LOAD_ASYNC_TO_LDS_B128 (GVS mode: saddr base + 32-bit vaddr byte offset).
// Tracked by ASYNCcnt; completion enforced by ASYNC_WAIT0() + workgroup barrier before use.
__device__ __forceinline__ void async_stage_chunk(const float* __restrict__ mu,
                                                  const float* __restrict__ Kf,
                                                  float* muc, float* kc,
                                                  int rbase, int tid) {
    const unsigned long long mub = (unsigned long long)(uintptr_t)mu;
    const unsigned long long kb  = (unsigned long long)(uintptr_t)Kf;
    #pragma unroll
    for (int it = 0; it < (NC * D_ / 4) / NTHREADS; ++it) {   // 16 iters of b128 per thread
        int idx = tid + NTHREADS * it;
        int rr  = idx >> 6;                 // row in chunk (64 float4 per row)
        int c4  = (idx & 63) * 4;           // float column
        // generic->LDS address: low 32 bits of a shared-memory pointer are the LDS offset
        unsigned lmu  = (unsigned)(uintptr_t)&muc[rr * LDP + c4];
        unsigned lk   = (unsigned)(uintptr_t)&kc [rr * LDP + c4];
        unsigned goff = (unsigned)(((rbase + rr) * D_ + c4) * 4);  // byte offset from base
        asm volatile("global_load_async_to_lds_b128 %0, %1, %2 offset:0"
                     :: "v"(lmu), "v"(goff), "s"(mub) : "memory");
        asm volatile("global_load_async_to_lds_b128 %0, %1, %2 offset:0"
                     :: "v"(lk), "v"(goff), "s"(kb) : "memory");
    }
}

// ---------------- main fused kernel ----------------
extern __shared__ float smem[];

__global__ __launch_bounds__(NTHREADS)
void gabor_main(const float* __restrict__ z, const float* __restrict__ mu,
                const float* __restrict__ Kf, const float* __restrict__ phi,
                const float* __restrict__ musq, const float* __restrict__ i2s,
                float* __restrict__ out) {
    float* zt   = smem;                    // [MT][LDP]
    float* muc  = zt  + MT * LDP;          // [NC][LDP]
    float* kc   = muc + NC * LDP;          // [NC][LDP]
    float* wc   = kc  + NC * LDP;          // [R_][WCP]  (un-normalized window*cos)
    float* accb = wc  + R_ * WCP;          // [8][256]   accumulator spill
    float* zsq  = accb + 8 * 256;          // [16]
    float* rsum = zsq + 16;                // [16]
    float* invs = rsum + 16;               // [16]

    const int tid  = threadIdx.x;
    const int lane = tid & 31;
    const int wv   = tid >> 5;
    const int row0 = blockIdx.x * MT;

    // kick off async DMA of the first mu/K chunk immediately
    async_stage_chunk(mu, Kf, muc, kc, 0, tid);

    // ---- stage z tile (16 x 256 f32), coalesced b128 ----
    #pragma unroll
    for (int it = 0; it < (MT * D_ / 4) / NTHREADS; ++it) {   // 4 iters
        int idx = tid + NTHREADS * it;        // float4 index
        int rr  = idx >> 6;                   // 64 float4 per row
        int c4  = (idx & 63) * 4;
        *(v4f*)&zt[rr * LDP + c4] = *(const v4f*)&z[(size_t)(row0 + rr) * D_ + c4];
    }
    if (tid < MT) rsum[tid] = 0.0f;
    __syncthreads();
    if (tid < MT) {                           // per-row |z|^2
        float s = 0.0f;
        for (int d = 0; d < D_; ++d) { float v = zt[tid * LDP + d]; s += v * v; }
        zsq[tid] = s;
    }

    // WMMA fragment addressing (f32 16x16x4, wave32):
    //  A: lanes 0-15 -> M rows, K pair {0,1}; lanes 16-31 -> K pair {2,3}
    //  B: lanes 0-15 -> N cols, K pair {0,1}; lanes 16-31 -> K pair {2,3}
    const int  mrow    = lane & 15;
    const int  khalf   = (lane >> 4) * 2;
    const int  tile    = wv & 3;              // which 16-wide N tile in the chunk
    const bool isPhase = (wv >= 4);           // waves 0-3: cross(mu), waves 4-7: phase(K)
    const int  m_e     = tid & 15;            // element-wise pass: fixed row per thread
    float wpart = 0.0f;                       // partial row sum of window

    for (int ch = 0; ch < R_ / NC; ++ch) {    // 16 chunks of 64 anchors
        const int rbase = ch * NC;

        // chunk data ready: each wave drains its own async ops, then barrier makes
        // every wave's slice visible to all waves
        ASYNC_WAIT0();
        __syncthreads();

        // ---- 8 waves = 4 N-tiles x {cross, phase}; 64 K-steps of fp32 WMMA ----
        const float* Ap = zt + mrow * LDP;
        const float* Bp = (isPhase ? kc : muc) + (tile * 16 + mrow) * LDP;
        v8f acc = {0.f, 0.f, 0.f, 0.f, 0.f, 0.f, 0.f, 0.f};
        #pragma unroll 8
        for (int k0 = 0; k0 < D_; k0 += 4) {
            v2f a = *(const v2f*)&Ap[k0 + khalf];
            v2f b = *(const v2f*)&Bp[k0 + khalf];
            acc = __builtin_amdgcn_wmma_f32_16x16x4_f32(
                      /*neg_a=*/false, a, /*neg_b=*/false, b,
                      /*c_mod=*/(short)0, acc, /*reuse_a=*/false, /*reuse_b=*/false);
        }
        // spill D-tile to LDS: C/D layout -> VGPR j, lane l: M = j + 8*(l>>4), N = l&15
        {
            float* dst = accb + wv * 256;
            const int hi = lane >> 4;
            #pragma unroll
            for (int j = 0; j < 8; ++j)
                dst[(j + 8 * hi) * 16 + (lane & 15)] = acc[j];
        }
        __syncthreads();   // all WMMA reads of muc/kc done; accb published

        // overlap next chunk's DMA with this chunk's transcendental-heavy pass
        if (ch + 1 < R_ / NC)
            async_stage_chunk(mu, Kf, muc, kc, rbase + NC, tid);

        // ---- element-wise: window, phase, scatter to wc, accumulate row sums ----
        #pragma unroll
        for (int i = 0; i < 4; ++i) {
            int rl = (tid >> 4) + 16 * i;     // 0..63 within chunk
            int tl = rl >> 4;
            int nn = rl & 15;
            int r  = rbase + rl;
            float cross = accb[tl * 256 + m_e * 16 + nn];
            float ph    = accb[(4 + tl) * 256 + m_e * 16 + nn] + phi[r];
            float dsq   = zsq[m_e] + musq[r] - 2.0f * cross;
            dsq = dsq > 0.0f ? dsq : 0.0f;
            float w = __expf(-dsq * i2s[r]);
            wpart += w;
            wc[r * WCP + m_e] = w * __cosf(ph);
        }
    }

    atomicAdd(&rsum[m_e], wpart);             // ds_add_f32, 16 lanes per address, once per WG
    __syncthreads();
    if (tid < MT) {
        float s = rsum[tid];
        invs[tid] = 1.0f / (s > 1e-6f ? s : 1e-6f);
    }
    __syncthreads();

    // ---- normalized write-out: fully coalesced over r ----
    for (int idx = tid; idx < MT * R_; idx += NTHREADS) {
        int mm = idx >> 10;                   // row within tile
        int r  = idx & (R_ - 1);
        out[(size_t)(row0 + mm) * R_ + r] = wc[r * WCP + mm] * invs[mm];
    }
}

// ---------------- host launcher ----------------
extern "C" void kernel_launch(void* const* d_in, const int* in_sizes, int n_in,
                              void* d_out, int out_size, void* d_ws, size_t ws_size,
                              hipStream_t stream) {
    const float* z   = (const float*)d_in[0];   // [B,T,D]
    const float* mu  = (const float*)d_in[1];   // [R,D]
    const float* Kf  = (const float*)d_in[2];   // [R,D]
    const float* ls  = (const float*)d_in[3];   // [R]
    const float* phi = (const float*)d_in[4];   // [R]
    float* out  = (float*)d_out;                // [B,T,R]
    float* musq = (float*)d_ws;                 // [R]
    float* i2s  = musq + R_;                    // [R]

    (void)in_sizes; (void)n_in; (void)out_size; (void)ws_size;

    hipFuncSetAttribute((const void*)gabor_main,
                        hipFuncAttributeMaxDynamicSharedMemorySize, SMEM_BYTES);

    gabor_precompute<<<(R_ * 32) / NTHREADS, NTHREADS, 0, stream>>>(mu, ls, musq, i2s);
    gabor_main<<<BT_ / MT, NTHREADS, SMEM_BYTES, stream>>>(z, mu, Kf, phi, musq, i2s, out);
}